// DeepGCN_40699110097665
// MI455X (gfx1250) — compile-verified
//
#include <hip/hip_runtime.h>
#include <hip/hip_bf16.h>
#include <math.h>

#define NN 50000
#define NE 800000
#define D  256
#define NCLS 7
#define NB  ((NN + 255) / 256)   // scan blocks = 196

typedef float v2f __attribute__((ext_vector_type(2)));
typedef float v8f __attribute__((ext_vector_type(8)));

// ---------------------------------------------------------------------------
// CSR construction (dst-sorted edge list) + normalization
// ---------------------------------------------------------------------------
__global__ void k_zero_int(int* __restrict__ p, int n) {
    int i = blockIdx.x * blockDim.x + threadIdx.x;
    if (i < n) p[i] = 0;
}

__global__ void k_count(const int* __restrict__ dst, int* __restrict__ cnt, int ne) {
    int e = blockIdx.x * blockDim.x + threadIdx.x;
    if (e < ne) atomicAdd(&cnt[dst[e]], 1);
}

__global__ void k_dinv(const int* __restrict__ cnt, float* __restrict__ dinv, int n) {
    int i = blockIdx.x * blockDim.x + threadIdx.x;
    if (i < n) dinv[i] = rsqrtf((float)(cnt[i] + 1));   // +1 self-loop
}

// block-local exclusive scan (Hillis-Steele, 256 wide) + block totals
__launch_bounds__(256)
__global__ void k_scan1(const int* __restrict__ cnt, int* __restrict__ rowtmp,
                        int* __restrict__ bsum, int n) {
    __shared__ int s[256];
    const int t = threadIdx.x;
    const int i = blockIdx.x * 256 + t;
    const int v = (i < n) ? cnt[i] : 0;
    s[t] = v;
    __syncthreads();
#pragma unroll
    for (int off = 1; off < 256; off <<= 1) {
        int x = (t >= off) ? s[t - off] : 0;
        __syncthreads();
        s[t] += x;
        __syncthreads();
    }
    if (i < n) rowtmp[i] = s[t] - v;          // exclusive within block
    if (t == 255) bsum[blockIdx.x] = s[255];  // block total
}

// single-block scan of the block totals
__launch_bounds__(256)
__global__ void k_scan2(const int* __restrict__ bsum, int* __restrict__ boff, int nb) {
    __shared__ int s[256];
    const int t = threadIdx.x;
    const int v = (t < nb) ? bsum[t] : 0;
    s[t] = v;
    __syncthreads();
#pragma unroll
    for (int off = 1; off < 256; off <<= 1) {
        int x = (t >= off) ? s[t - off] : 0;
        __syncthreads();
        s[t] += x;
        __syncthreads();
    }
    if (t < nb) boff[t] = s[t] - v;           // exclusive block offsets
}

__global__ void k_scan3(const int* __restrict__ rowtmp, const int* __restrict__ boff,
                        int* __restrict__ row_ptr, int n, int ne) {
    int i = blockIdx.x * blockDim.x + threadIdx.x;
    if (i < n) row_ptr[i] = rowtmp[i] + boff[i >> 8];
    if (i == 0) row_ptr[n] = ne;
}

// drop each edge into its CSR slot; fold norm = dinv[src]*dinv[dst] into slot
__global__ void k_fill(const int* __restrict__ src, const int* __restrict__ dst,
                       const float* __restrict__ dinv, const int* __restrict__ row_ptr,
                       int* __restrict__ fill, int* __restrict__ srcp,
                       float* __restrict__ normp, int ne) {
    int e = blockIdx.x * blockDim.x + threadIdx.x;
    if (e < ne) {
        const int d = dst[e];
        const int s = src[e];
        const int pos  = atomicAdd(&fill[d], 1);
        const int slot = row_ptr[d] + pos;
        srcp[slot]  = s;
        normp[slot] = dinv[s] * dinv[d];
    }
}

// ---------------------------------------------------------------------------
// H[n,256] = X[n,256] @ W[256,256]   (f32 WMMA 16x16x4, wave32)
// block = 256 threads = 8 waves, one 16-row strip / block, 2 col tiles / wave
// ---------------------------------------------------------------------------
__launch_bounds__(256)
__global__ void k_gemm256(const float* __restrict__ X, const float* __restrict__ W,
                          float* __restrict__ H) {
    __shared__ __align__(16) float As[16 * 260];   // stride 260: bank-conflict-free
    const int row0 = blockIdx.x * 16;
    const int t = threadIdx.x;
    {
        const int r  = t >> 4;
        const int c0 = (t & 15) << 4;
        const float4* s4 = (const float4*)(X + (size_t)(row0 + r) * D + c0);
        float4* d4 = (float4*)&As[r * 260 + c0];
        d4[0] = s4[0]; d4[1] = s4[1]; d4[2] = s4[2]; d4[3] = s4[3];
    }
    __syncthreads();

    const int lane = t & 31;
    const int wave = t >> 5;
    const int m    = lane & 15;          // M index for A, N index for B/D
    const int koff = (lane >> 4) << 1;   // lane half selects K pair: 0 or 2
    const int c0   = wave * 32;
    const int c1   = c0 + 16;

    v8f acc0 = {};
    v8f acc1 = {};
    for (int k0 = 0; k0 < D; k0 += 4) {
        const v2f a = *(const v2f*)&As[m * 260 + k0 + koff];
        const float* wp = W + (size_t)(k0 + koff) * D;
        v2f b0; b0.x = wp[c0 + m]; b0.y = wp[D + c0 + m];
        v2f b1; b1.x = wp[c1 + m]; b1.y = wp[D + c1 + m];
        acc0 = __builtin_amdgcn_wmma_f32_16x16x4_f32(false, a, false, b0, (short)0, acc0, false, false);
        acc1 = __builtin_amdgcn_wmma_f32_16x16x4_f32(false, a, false, b1, (short)0, acc1, false, false);
    }

    const int mrow = (lane >> 4) << 3;
#pragma unroll
    for (int r = 0; r < 8; ++r) {
        H[(size_t)(row0 + mrow + r) * D + c0 + m] = acc0[r];
        H[(size_t)(row0 + mrow + r) * D + c1 + m] = acc1[r];
    }
}

// ---------------------------------------------------------------------------
// P[n,8] = h2[n,256] @ W2p[256,16] (cols 7..15 zero-padded; keep first 8)
// ---------------------------------------------------------------------------
__launch_bounds__(32)
__global__ void k_gemm_out(const float* __restrict__ Hsrc, const float* __restrict__ W2p,
                           float* __restrict__ P) {
    __shared__ __align__(16) float As[16 * 260];
    const int row0 = blockIdx.x * 16;
    const int t = threadIdx.x;
    {
        const int r = t >> 1;
        const int h = (t & 1) * 128;
        const float4* s4 = (const float4*)(Hsrc + (size_t)(row0 + r) * D + h);
        float4* d4 = (float4*)&As[r * 260 + h];
#pragma unroll
        for (int i = 0; i < 32; ++i) d4[i] = s4[i];
    }
    __syncthreads();

    const int m    = t & 15;
    const int koff = (t >> 4) << 1;
    v8f acc = {};
    for (int k0 = 0; k0 < D; k0 += 4) {
        const v2f a = *(const v2f*)&As[m * 260 + k0 + koff];
        const float* wp = W2p + (size_t)(k0 + koff) * 16;
        v2f b; b.x = wp[m]; b.y = wp[16 + m];
        acc = __builtin_amdgcn_wmma_f32_16x16x4_f32(false, a, false, b, (short)0, acc, false, false);
    }
    if (m < 8) {
        const int mrow = (t >> 4) << 3;
#pragma unroll
        for (int r = 0; r < 8; ++r)
            P[(size_t)(row0 + mrow + r) * 8 + m] = acc[r];
    }
}

// ---------------------------------------------------------------------------
// CSR gather aggregation, 256-wide: one wave per node, no atomics.
// lane holds 8 features (2 float4): [lane*4, lane*4+4) and [128+lane*4, ...)
// ---------------------------------------------------------------------------
__device__ __forceinline__ void fma4(float4& a, const float4 h, const float c) {
    a.x += h.x * c; a.y += h.y * c; a.z += h.z * c; a.w += h.w * c;
}

__launch_bounds__(256)
__global__ void k_aggregate(const float* __restrict__ H, const float* __restrict__ dinv,
                            const int* __restrict__ row_ptr, const int* __restrict__ srcp,
                            const float* __restrict__ normp, float* __restrict__ A) {
    const int t    = threadIdx.x;
    const int lane = t & 31;
    const int node = blockIdx.x * 8 + (t >> 5);

    const float di = dinv[node];
    const float s2 = di * di;

    const float4* hp = (const float4*)(H + (size_t)node * D);
    float4 a0 = hp[lane];
    float4 a1 = hp[lane + 32];
    a0.x *= s2; a0.y *= s2; a0.z *= s2; a0.w *= s2;   // self-loop term
    a1.x *= s2; a1.y *= s2; a1.z *= s2; a1.w *= s2;

    int j = row_ptr[node];
    const int end = row_ptr[node + 1];
    for (; j + 2 <= end; j += 2) {                    // 2-way unroll for ILP
        const int   sA = srcp[j],     sB = srcp[j + 1];
        const float cA = normp[j],    cB = normp[j + 1];
        const float4* pA = (const float4*)(H + (size_t)sA * D);
        const float4* pB = (const float4*)(H + (size_t)sB * D);
        const float4 hA0 = pA[lane], hA1 = pA[lane + 32];
        const float4 hB0 = pB[lane], hB1 = pB[lane + 32];
        fma4(a0, hA0, cA); fma4(a1, hA1, cA);
        fma4(a0, hB0, cB); fma4(a1, hB1, cB);
    }
    if (j < end) {
        const int   s = srcp[j];
        const float c = normp[j];
        const float4* p = (const float4*)(H + (size_t)s * D);
        fma4(a0, p[lane], c);
        fma4(a1, p[lane + 32], c);
    }

    float4* ap = (float4*)(A + (size_t)node * D);
    ap[lane]      = a0;
    ap[lane + 32] = a1;
}

// 8-wide CSR gather aggregation for the output layer (8 threads per node)
__launch_bounds__(256)
__global__ void k_aggregate8(const float* __restrict__ P, const float* __restrict__ dinv,
                             const int* __restrict__ row_ptr, const int* __restrict__ srcp,
                             const float* __restrict__ normp, float* __restrict__ Pagg, int n) {
    const int t = threadIdx.x;
    const int node = blockIdx.x * 32 + (t >> 3);
    if (node >= n) return;
    const int c = t & 7;
    const float di = dinv[node];
    float acc = P[(size_t)node * 8 + c] * di * di;
    const int end = row_ptr[node + 1];
    for (int j = row_ptr[node]; j < end; ++j)
        acc += P[(size_t)srcp[j] * 8 + c] * normp[j];
    Pagg[(size_t)node * 8 + c] = acc;
}

// ---------------------------------------------------------------------------
// pointwise epilogues
// ---------------------------------------------------------------------------
__global__ void k_bias_relu(const float* __restrict__ A, const float* __restrict__ b,
                            float* __restrict__ out, int n4) {
    int i = blockIdx.x * blockDim.x + threadIdx.x;
    if (i < n4) {
        float4 v = ((const float4*)A)[i];
        float4 bb = ((const float4*)b)[i & 63];
        v.x = fmaxf(v.x + bb.x, 0.0f); v.y = fmaxf(v.y + bb.y, 0.0f);
        v.z = fmaxf(v.z + bb.z, 0.0f); v.w = fmaxf(v.w + bb.w, 0.0f);
        ((float4*)out)[i] = v;
    }
}

__global__ void k_bias_relu_res(float* __restrict__ A, const float* __restrict__ b,
                                const float* __restrict__ h1, int n4) {
    int i = blockIdx.x * blockDim.x + threadIdx.x;
    if (i < n4) {
        float4 v = ((const float4*)A)[i];
        float4 bb = ((const float4*)b)[i & 63];
        float4 r = ((const float4*)h1)[i];
        v.x = fmaxf(v.x + bb.x, 0.0f) + r.x; v.y = fmaxf(v.y + bb.y, 0.0f) + r.y;
        v.z = fmaxf(v.z + bb.z, 0.0f) + r.z; v.w = fmaxf(v.w + bb.w, 0.0f) + r.w;
        ((float4*)A)[i] = v;
    }
}

__global__ void k_pad_w2(const float* __restrict__ W2, float* __restrict__ W2p) {
    int i = blockIdx.x * blockDim.x + threadIdx.x;
    if (i < 256 * 16) {
        int k = i >> 4, c = i & 15;
        W2p[i] = (c < NCLS) ? W2[k * NCLS + c] : 0.0f;
    }
}

__global__ void k_logsoftmax(const float* __restrict__ Pagg, const float* __restrict__ b2,
                             float* __restrict__ out, int n) {
    int i = blockIdx.x * blockDim.x + threadIdx.x;
    if (i < n) {
        float v[NCLS];
        float mx = -INFINITY;
#pragma unroll
        for (int c = 0; c < NCLS; ++c) {
            v[c] = Pagg[(size_t)i * 8 + c] + b2[c];
            mx = fmaxf(mx, v[c]);
        }
        float s = 0.0f;
#pragma unroll
        for (int c = 0; c < NCLS; ++c) s += expf(v[c] - mx);
        const float ls = logf(s) + mx;
#pragma unroll
        for (int c = 0; c < NCLS; ++c) out[(size_t)i * NCLS + c] = v[c] - ls;
    }
}

// ---------------------------------------------------------------------------
extern "C" void kernel_launch(void* const* d_in, const int* in_sizes, int n_in,
                              void* d_out, int out_size, void* d_ws, size_t ws_size,
                              hipStream_t stream) {
    const float* x  = (const float*)d_in[0];
    const int*   ei = (const int*)  d_in[1];
    const float* W0 = (const float*)d_in[2];
    const float* b0 = (const float*)d_in[3];
    const float* W1 = (const float*)d_in[4];
    const float* b1 = (const float*)d_in[5];
    const float* W2 = (const float*)d_in[6];
    const float* b2 = (const float*)d_in[7];
    const int* src = ei;
    const int* dst = ei + NE;
    float* out = (float*)d_out;

    // workspace carve-up (256B aligned chunks)
    char* base = (char*)d_ws;
    auto carve = [&](size_t bytes) -> void* {
        void* p = (void*)base;
        base += (bytes + 255) & ~(size_t)255;
        return p;
    };
    float* dinv    = (float*)carve(NN * 4);
    int*   cnt     = (int*)  carve(NN * 4);          // degree count, reused as fill ctr
    int*   rowtmp  = (int*)  carve(NN * 4);
    int*   bsum    = (int*)  carve(256 * 4);
    int*   boff    = (int*)  carve(256 * 4);
    int*   row_ptr = (int*)  carve((NN + 1) * 4);
    int*   srcp    = (int*)  carve((size_t)NE * 4);  // CSR-permuted sources
    float* normp   = (float*)carve((size_t)NE * 4);  // CSR-permuted edge norms
    float* H       = (float*)carve((size_t)NN * D * 4);
    float* A       = (float*)carve((size_t)NN * D * 4);
    float* h1      = (float*)carve((size_t)NN * D * 4);
    float* P       = (float*)carve((size_t)NN * 8 * 4);
    float* Pagg    = (float*)carve((size_t)NN * 8 * 4);
    float* W2p     = (float*)carve(256 * 16 * 4);

    const int n4 = NN * 64;   // #float4 per 256-wide tensor

    // ---- CSR build + normalization (shared by all layers) ----
    k_zero_int<<<NB, 256, 0, stream>>>(cnt, NN);
    k_count   <<<NE / 256, 256, 0, stream>>>(dst, cnt, NE);
    k_dinv    <<<NB, 256, 0, stream>>>(cnt, dinv, NN);
    k_scan1   <<<NB, 256, 0, stream>>>(cnt, rowtmp, bsum, NN);
    k_scan2   <<<1, 256, 0, stream>>>(bsum, boff, NB);
    k_scan3   <<<NB, 256, 0, stream>>>(rowtmp, boff, row_ptr, NN, NE);
    k_zero_int<<<NB, 256, 0, stream>>>(cnt, NN);     // reuse as fill counter
    k_fill    <<<NE / 256, 256, 0, stream>>>(src, dst, dinv, row_ptr, cnt, srcp, normp, NE);
    k_pad_w2  <<<16, 256, 0, stream>>>(W2, W2p);

    // ---- layer 1: h1 = relu(agg(x @ W0) + b0) ----
    k_gemm256  <<<NN / 16, 256, 0, stream>>>(x, W0, H);
    k_aggregate<<<NN / 8, 256, 0, stream>>>(H, dinv, row_ptr, srcp, normp, A);
    k_bias_relu<<<(n4 + 255) / 256, 256, 0, stream>>>(A, b0, h1, n4);

    // ---- layer 2: h2 = relu(agg(h1 @ W1) + b1) + h1   (h2 lands in A) ----
    k_gemm256      <<<NN / 16, 256, 0, stream>>>(h1, W1, H);
    k_aggregate    <<<NN / 8, 256, 0, stream>>>(H, dinv, row_ptr, srcp, normp, A);
    k_bias_relu_res<<<(n4 + 255) / 256, 256, 0, stream>>>(A, b1, h1, n4);

    // ---- output: log_softmax(agg(h2 @ W2) + b2) ----
    k_gemm_out  <<<NN / 16, 32, 0, stream>>>(A, W2p, P);
    k_aggregate8<<<(NN + 31) / 32, 256, 0, stream>>>(P, dinv, row_ptr, srcp, normp, Pagg, NN);
    k_logsoftmax<<<NB, 256, 0, stream>>>(Pagg, b2, out, NN);
}